// GAPLayer_single_80049600462883
// MI455X (gfx1250) — compile-verified
//
#include <hip/hip_runtime.h>
#include <math.h>

// ---------------- problem constants (from reference setup_inputs) ----------------
#define EPSF  1e-5f
#define SLOPE 0.2f
constexpr int  Bn   = 16;
constexpr int  Nn   = 8192;
constexpr int  Kn   = 20;
constexpr int  CHn  = 16;
constexpr long long En  = (long long)Bn * Nn * Kn;   // 2,621,440 edge elements
constexpr int  NKn  = Nn * Kn;                        // 163,840
constexpr int  ROWS = Bn * Nn;                        // 131,072
constexpr long long OUT0 = (long long)ROWS * CHn;     // offset of edge_o_p in d_out

typedef __attribute__((ext_vector_type(2))) float v2f;
typedef __attribute__((ext_vector_type(4))) float v4f;
typedef __attribute__((ext_vector_type(8))) float v8f;

// ---------------- ws layout (bytes) ----------------
// 0     : double acc[16]   ([0..4] pass1 moments Sd,Sd2,Sx,Sx2,Sdx ; [8..11] pass2 Sa,Sa2,Se,Se2)
// 128   : float  c1[96]    (ad1[16], ax1[16], bt1[16], ad1n[16], ax1n[16], bt1n[16])
// 512   : float  c2[4]     (ca, ce, c0)
// 1024  : float  d_arr[E]
// +4E   : float  sa[E]
// +8E   : float  se[E]

__device__ __forceinline__ float lrelu(float v) { return v > 0.0f ? v : SLOPE * v; }

__global__ void k_init(double* acc) {
    if (threadIdx.x < 16) acc[threadIdx.x] = 0.0;
}

// ---------------- pass 1: gather, d = x[idx] - x[n], global moments ----------------
__global__ void k_pass1(const float* __restrict__ x, const int* __restrict__ idx,
                        double* __restrict__ acc, float* __restrict__ d_arr)
{
    long long tid    = (long long)blockIdx.x * blockDim.x + threadIdx.x;
    long long stride = (long long)gridDim.x * blockDim.x;
    float sd = 0.f, sd2 = 0.f, sx = 0.f, sx2 = 0.f, sdx = 0.f;
    for (long long e = tid; e < En; e += stride) {
        int b = (int)(e / NKn);
        int n = (int)((e % NKn) / Kn);
        int j = idx[e];
        float xc = x[b * Nn + n];
        float g  = x[b * Nn + j];
        float d  = g - xc;
        d_arr[e] = d;
        sd += d; sd2 += d * d; sx += xc; sx2 += xc * xc; sdx += d * xc;
    }
    #pragma unroll
    for (int off = 16; off >= 1; off >>= 1) {
        sd  += __shfl_xor(sd,  off, 32);
        sd2 += __shfl_xor(sd2, off, 32);
        sx  += __shfl_xor(sx,  off, 32);
        sx2 += __shfl_xor(sx2, off, 32);
        sdx += __shfl_xor(sdx, off, 32);
    }
    if ((threadIdx.x & 31) == 0) {
        atomicAdd(&acc[0], (double)sd);
        atomicAdd(&acc[1], (double)sd2);
        atomicAdd(&acc[2], (double)sx);
        atomicAdd(&acc[3], (double)sx2);
        atomicAdd(&acc[4], (double)sdx);
    }
}

// ---------------- stats 1: fold BN1 into affine per channel ----------------
__global__ void k_stats1(const double* __restrict__ acc,
                         const float* __restrict__ W1, const float* __restrict__ g1, const float* __restrict__ b1,
                         const float* __restrict__ W1n, const float* __restrict__ g1n, const float* __restrict__ b1n,
                         float* __restrict__ c1)
{
    int t = threadIdx.x;
    if (t >= 32) return;
    int  c      = t & 15;
    bool second = t >= 16;
    const float* W  = second ? W1n : W1;
    const float* gv = second ? g1n : g1;
    const float* bv = second ? b1n : b1;
    double invE = 1.0 / (double)En;
    double m0 = acc[0] * invE, q0  = acc[1] * invE;
    double m1 = acc[2] * invE, q1  = acc[3] * invE;
    double q01 = acc[4] * invE;
    double w0 = (double)W[c * 2 + 0], w1 = (double)W[c * 2 + 1];
    double mean = w0 * m0 + w1 * m1;
    double ez2  = w0 * w0 * q0 + 2.0 * w0 * w1 * q01 + w1 * w1 * q1;
    double var  = ez2 - mean * mean;
    float  r    = (float)(1.0 / sqrt(var + (double)EPSF));
    float  gg   = gv[c];
    int base = second ? 48 : 0;
    c1[base + c]      = gg * (float)w0 * r;            // alpha_d
    c1[base + 16 + c] = gg * (float)w1 * r;            // alpha_x
    c1[base + 32 + c] = bv[c] - gg * (float)mean * r;  // beta
}

// ---------------- pass 2: WMMA conv1+BN1 affine, lrelu, conv2 dots, edge_o_p ----------------
__global__ void k_pass2(const float* __restrict__ x, const float* __restrict__ d_arr,
                        const float* __restrict__ c1,
                        const float* __restrict__ W2, const float* __restrict__ W2n,
                        float* __restrict__ edge_out, float* __restrict__ sa_arr,
                        float* __restrict__ se_arr, double* __restrict__ acc)
{
    int  lid = threadIdx.x & 31;
    int  jj  = lid & 15;
    bool hi  = lid >= 16;

    // A matrix (16x4 f32): lanes 0-15 hold K=0 (VGPR0) / K=1 (VGPR1); lanes 16-31 hold K=2 / K=3
    v2f A, An;
    A[0]  = hi ? c1[32 + jj] : c1[jj];        // beta      : alpha_d
    A[1]  = hi ? 0.0f        : c1[16 + jj];   // 0         : alpha_x
    An[0] = hi ? c1[80 + jj] : c1[48 + jj];
    An[1] = hi ? 0.0f        : c1[64 + jj];

    float w2v[8], w2nv[8];
    #pragma unroll
    for (int r = 0; r < 8; r++) {
        w2v[r]  = W2 [(hi ? 8 : 0) + r];
        w2nv[r] = W2n[(hi ? 8 : 0) + r];
    }

    long long wave   = ((long long)blockIdx.x * blockDim.x + threadIdx.x) >> 5;
    long long nwaves = ((long long)gridDim.x * blockDim.x) >> 5;
    const long long NGRP = En / 16;

    float aS = 0.f, aS2 = 0.f, eS = 0.f, eS2 = 0.f;

    for (long long grp = wave; grp < NGRP; grp += nwaves) {
        long long e = grp * 16 + jj;
        int b = (int)(e / NKn);
        int n = (int)((e % NKn) / Kn);
        float dval = d_arr[e];
        float xc   = x[b * Nn + n];

        // B matrix (4x16): column j = [d, xc, 1, 0]^T
        v2f Bv;
        Bv[0] = hi ? 1.0f : dval;
        Bv[1] = hi ? 0.0f : xc;

        v8f Cz = {};
        v8f z1 = __builtin_amdgcn_wmma_f32_16x16x4_f32(false, A,  false, Bv, (short)0, Cz, false, false);
        v8f z2 = __builtin_amdgcn_wmma_f32_16x16x4_f32(false, An, false, Bv, (short)0, Cz, false, false);

        float eo[8], pa = 0.f, pe = 0.f;
        #pragma unroll
        for (int r = 0; r < 8; r++) {
            float hv = lrelu(z1[r]);
            float ev = lrelu(z2[r]);
            eo[r] = ev;
            pa += w2v[r] * hv;
            pe += w2nv[r] * ev;
        }

        // edge_o_p store: element e, channels (hi?8:0)+0..7, layout e*16 + c
        float* ep = edge_out + e * 16 + (hi ? 8 : 0);
        v4f s0 = { eo[0], eo[1], eo[2], eo[3] };
        v4f s1 = { eo[4], eo[5], eo[6], eo[7] };
        *(v4f*)(ep)     = s0;
        *(v4f*)(ep + 4) = s1;

        float sa = pa + __shfl_xor(pa, 16, 32);
        float se = pe + __shfl_xor(pe, 16, 32);
        if (!hi) {
            sa_arr[e] = sa;
            se_arr[e] = se;
            aS += sa; aS2 += sa * sa; eS += se; eS2 += se * se;
        }
    }
    #pragma unroll
    for (int off = 16; off >= 1; off >>= 1) {
        aS  += __shfl_xor(aS,  off, 32);
        aS2 += __shfl_xor(aS2, off, 32);
        eS  += __shfl_xor(eS,  off, 32);
        eS2 += __shfl_xor(eS2, off, 32);
    }
    if (lid == 0) {
        atomicAdd(&acc[8],  (double)aS);
        atomicAdd(&acc[9],  (double)aS2);
        atomicAdd(&acc[10], (double)eS);
        atomicAdd(&acc[11], (double)eS2);
    }
}

// ---------------- stats 2: BN2 constants ----------------
__global__ void k_stats2(const double* __restrict__ acc,
                         const float* __restrict__ g2, const float* __restrict__ b2,
                         const float* __restrict__ g2n, const float* __restrict__ b2n,
                         float* __restrict__ c2)
{
    if (threadIdx.x != 0) return;
    double invE = 1.0 / (double)En;
    double ma = acc[8] * invE,  va = acc[9]  * invE - ma * ma;
    double me = acc[10] * invE, ve = acc[11] * invE - me * me;
    float ca = g2[0]  * (float)(1.0 / sqrt(va + (double)EPSF));
    float ce = g2n[0] * (float)(1.0 / sqrt(ve + (double)EPSF));
    float c0 = (b2[0] - ca * (float)ma) + (b2n[0] - ce * (float)me);
    c2[0] = ca; c2[1] = ce; c2[2] = c0;
}

// ---------------- pass 3: softmax over K=20 + weighted sum ----------------
__global__ void k_pass3(const float* __restrict__ sa_arr, const float* __restrict__ se_arr,
                        const float* __restrict__ c2, const float* __restrict__ edge,
                        float* __restrict__ out)
{
    int r = blockIdx.x * blockDim.x + threadIdx.x;
    if (r >= ROWS) return;
    float ca = c2[0], ce = c2[1], c0 = c2[2];
    long long base = (long long)r * Kn;

    float t[Kn];
    float m = -1e30f;
    #pragma unroll
    for (int k = 0; k < Kn; k++) {
        float v = lrelu(ca * sa_arr[base + k] + ce * se_arr[base + k] + c0);
        t[k] = v;
        m = fmaxf(m, v);
    }
    float s = 0.f;
    #pragma unroll
    for (int k = 0; k < Kn; k++) { t[k] = __expf(t[k] - m); s += t[k]; }
    float inv = 1.0f / s;

    v4f o0 = {}, o1 = {}, o2 = {}, o3 = {};
    #pragma unroll
    for (int k = 0; k < Kn; k++) {
        float w = t[k] * inv;
        const v4f* ep = (const v4f*)(edge + (base + k) * 16);
        o0 += w * ep[0];
        o1 += w * ep[1];
        o2 += w * ep[2];
        o3 += w * ep[3];
    }
    v4f* op = (v4f*)(out + (long long)r * 16);
    op[0] = o0; op[1] = o1; op[2] = o2; op[3] = o3;
}

// ---------------- launcher ----------------
extern "C" void kernel_launch(void* const* d_in, const int* in_sizes, int n_in,
                              void* d_out, int out_size, void* d_ws, size_t ws_size,
                              hipStream_t stream)
{
    const float* x   = (const float*)d_in[0];
    // d_in[1] = pos (unused by reference), d_in[3] = dis (unused by reference)
    const int*   idx = (const int*)d_in[2];
    const float* W1  = (const float*)d_in[4];
    const float* g1  = (const float*)d_in[5];
    const float* b1  = (const float*)d_in[6];
    const float* W2  = (const float*)d_in[7];
    const float* g2  = (const float*)d_in[8];
    const float* b2  = (const float*)d_in[9];
    const float* W1n = (const float*)d_in[10];
    const float* g1n = (const float*)d_in[11];
    const float* b1n = (const float*)d_in[12];
    const float* W2n = (const float*)d_in[13];
    const float* g2n = (const float*)d_in[14];
    const float* b2n = (const float*)d_in[15];

    double* acc   = (double*)d_ws;
    float*  c1    = (float*)((char*)d_ws + 128);
    float*  c2    = (float*)((char*)d_ws + 512);
    float*  d_arr = (float*)((char*)d_ws + 1024);
    float*  sa    = d_arr + En;
    float*  se    = sa + En;

    float* out  = (float*)d_out;
    float* edge = out + OUT0;

    k_init  <<<1, 32, 0, stream>>>(acc);
    k_pass1 <<<1024, 256, 0, stream>>>(x, idx, acc, d_arr);
    k_stats1<<<1, 32, 0, stream>>>(acc, W1, g1, b1, W1n, g1n, b1n, c1);
    k_pass2 <<<512, 256, 0, stream>>>(x, d_arr, c1, W2, W2n, edge, sa, se, acc);
    k_stats2<<<1, 32, 0, stream>>>(acc, g2, b2, g2n, b2n, c2);
    k_pass3 <<<(ROWS + 255) / 256, 256, 0, stream>>>(sa, se, c2, edge, out);
}